// GraphConvBlock_23141283791389
// MI455X (gfx1250) — compile-verified
//
#include <hip/hip_runtime.h>

#define CCH 128  // channels

typedef float v2f __attribute__((ext_vector_type(2)));
typedef float v8f __attribute__((ext_vector_type(8)));

// ---------------- degree (in-degree over edges; self-loop folded as +1 later) ----------------
__global__ __launch_bounds__(256) void deg_kernel(const int* __restrict__ dst,
                                                  float* __restrict__ deg, int E) {
  int i = blockIdx.x * blockDim.x + threadIdx.x;
  if (i < E) atomicAdd(&deg[dst[i]], 1.0f);
}

__global__ __launch_bounds__(256) void dinv_kernel(const float* __restrict__ deg,
                                                   float* __restrict__ dinv, int N) {
  int i = blockIdx.x * blockDim.x + threadIdx.x;
  if (i < N) dinv[i] = rsqrtf(deg[i] + 1.0f);  // +1 = self loop; always > 0
}

// ---------------- h = x @ W via V_WMMA_F32_16X16X4_F32 ----------------
// One block = 256 threads = 8 waves. Wave w computes the 16x16 tile
// (rows blockIdx.x*16 .. +15) x (cols w*16 .. +15). K loop = 128/4 = 32 WMMAs.
__global__ __launch_bounds__(256) void gemm_wmma_kernel(const float* __restrict__ x,
                                                        const float* __restrict__ W,
                                                        float* __restrict__ h, int N) {
  const int lane = threadIdx.x & 31;
  const int wave = threadIdx.x >> 5;     // 0..7 -> column tile
  const int m    = lane & 15;
  const int hi   = lane >> 4;            // half-wave select
  const int row0 = blockIdx.x * 16;
  const int col0 = wave * 16;

  int rowA = row0 + m;
  if (rowA >= N) rowA = N - 1;           // clamp loads; keeps EXEC all-ones for WMMA
  const float* __restrict__ arow = x + (size_t)rowA * CCH + 2 * hi;

  v8f acc = {};
  for (int k = 0; k < CCH; k += 4) {
    // A 16x4 f32: VGPR0 = K = 2*hi, VGPR1 = K = 2*hi+1  (contiguous 8B load)
    v2f a = *(const v2f*)(arow + k);
    // B 4x16 f32: VGPR0 = row K=2*hi, VGPR1 = row K=2*hi+1, N = lane%16
    v2f b;
    b.x = W[(size_t)(k + 2 * hi + 0) * CCH + col0 + m];
    b.y = W[(size_t)(k + 2 * hi + 1) * CCH + col0 + m];
    acc = __builtin_amdgcn_wmma_f32_16x16x4_f32(false, a, false, b,
                                                (short)0, acc, false, false);
  }
  // D 16x16 f32: VGPR v -> M = v + 8*hi, N = lane%16
#pragma unroll
  for (int v = 0; v < 8; ++v) {
    int row = row0 + v + 8 * hi;
    if (row < N) h[(size_t)row * CCH + col0 + m] = acc[v];
  }
}

// ---------------- edge scatter: agg[dst] += h[src] * dinv[src]*dinv[dst] ----------------
// One wave per edge; each lane gathers float4 (coalesced 512B/wave) and issues 4 atomics.
// agg + h fit in the 192MB L2, so the atomic RMWs stay on-die.
__global__ __launch_bounds__(256) void scatter_kernel(const int* __restrict__ src,
                                                      const int* __restrict__ dst,
                                                      const float* __restrict__ dinv,
                                                      const float* __restrict__ h,
                                                      float* __restrict__ agg, int E) {
  int e = blockIdx.x * 8 + (threadIdx.x >> 5);
  if (e >= E) return;
  int lane = threadIdx.x & 31;
  int s = src[e], d = dst[e];
  float nrm = dinv[s] * dinv[d];
  const float4 hv = *(const float4*)(h + (size_t)s * CCH + lane * 4);
  float* ap = agg + (size_t)d * CCH + lane * 4;
  atomicAdd(ap + 0, hv.x * nrm);
  atomicAdd(ap + 1, hv.y * nrm);
  atomicAdd(ap + 2, hv.z * nrm);
  atomicAdd(ap + 3, hv.w * nrm);
}

// ---------------- fuse self-loop + bias, accumulate per-channel BN statistics ----------------
// blockDim = 128 (one thread per channel), 64 rows per block, 2 atomics/thread.
__global__ __launch_bounds__(128) void stats_kernel(float* __restrict__ agg,
                                                    const float* __restrict__ h,
                                                    const float* __restrict__ dinv,
                                                    const float* __restrict__ bias,
                                                    float* __restrict__ sums,
                                                    float* __restrict__ sumsq, int N) {
  int c = threadIdx.x;
  int row0 = blockIdx.x * 64;
  int rmax = N - row0; if (rmax > 64) rmax = 64;
  float bc = bias[c];
  float s = 0.0f, ss = 0.0f;
  for (int r = 0; r < rmax; ++r) {
    int i = row0 + r;
    float di = dinv[i];
    size_t off = (size_t)i * CCH + c;
    float v = agg[off] + h[off] * (di * di) + bc;  // add self-loop msg + bias
    agg[off] = v;
    s += v; ss += v * v;
  }
  atomicAdd(&sums[c], s);
  atomicAdd(&sumsq[c], ss);
}

// ---------------- BatchNorm (batch stats) + ReLU + residual ----------------
__global__ __launch_bounds__(256) void final_kernel(const float* __restrict__ agg,
                                                    const float* __restrict__ x,
                                                    const float* __restrict__ gamma,
                                                    const float* __restrict__ beta,
                                                    const float* __restrict__ sums,
                                                    const float* __restrict__ sumsq,
                                                    float* __restrict__ out,
                                                    int N, long long total) {
  long long idx = (long long)blockIdx.x * blockDim.x + threadIdx.x;
  if (idx >= total) return;
  int c = (int)(idx & (CCH - 1));
  float invN = 1.0f / (float)N;
  float mean = sums[c] * invN;
  float var  = sumsq[c] * invN - mean * mean;   // biased variance
  float v = gamma[c] * (agg[idx] - mean) * rsqrtf(var + 1e-5f) + beta[c];
  v = v > 0.0f ? v : 0.0f;
  out[idx] = v + x[idx];
}

extern "C" void kernel_launch(void* const* d_in, const int* in_sizes, int n_in,
                              void* d_out, int out_size, void* d_ws, size_t ws_size,
                              hipStream_t stream) {
  const float* x     = (const float*)d_in[0];
  const float* W     = (const float*)d_in[1];
  const float* bias  = (const float*)d_in[2];
  const float* gamma = (const float*)d_in[3];
  const float* beta  = (const float*)d_in[4];
  const int*   ei    = (const int*)d_in[5];

  const int N = in_sizes[0] / CCH;
  const int E = in_sizes[5] / 2;
  const int* src = ei;        // edge_index[0] = message sources
  const int* dst = ei + E;    // edge_index[1] = aggregation targets

  // Workspace carve-out (256B aligned slices)
  char* ws = (char*)d_ws;
  size_t off = 0;
  auto carve = [&](size_t bytes) -> float* {
    float* p = (float*)(ws + off);
    off += (bytes + 255) & ~(size_t)255;
    return p;
  };
  float* deg   = carve((size_t)N * sizeof(float));
  float* dinv  = carve((size_t)N * sizeof(float));
  float* h     = carve((size_t)N * CCH * sizeof(float));
  float* agg   = carve((size_t)N * CCH * sizeof(float));
  float* sums  = carve(CCH * sizeof(float));
  float* sumsq = carve(CCH * sizeof(float));

  // Zero accumulators (graph-capture-safe async memsets, stream ordered)
  hipMemsetAsync(deg,   0, (size_t)N * sizeof(float), stream);
  hipMemsetAsync(agg,   0, (size_t)N * CCH * sizeof(float), stream);
  hipMemsetAsync(sums,  0, CCH * sizeof(float), stream);
  hipMemsetAsync(sumsq, 0, CCH * sizeof(float), stream);

  deg_kernel<<<(E + 255) / 256, 256, 0, stream>>>(dst, deg, E);
  dinv_kernel<<<(N + 255) / 256, 256, 0, stream>>>(deg, dinv, N);
  gemm_wmma_kernel<<<(N + 15) / 16, 256, 0, stream>>>(x, W, h, N);
  scatter_kernel<<<(E + 7) / 8, 256, 0, stream>>>(src, dst, dinv, h, agg, E);
  stats_kernel<<<(N + 63) / 64, 128, 0, stream>>>(agg, h, dinv, bias, sums, sumsq, N);

  long long total = (long long)N * CCH;
  final_kernel<<<(unsigned)((total + 255) / 256), 256, 0, stream>>>(
      agg, x, gamma, beta, sums, sumsq, (float*)d_out, N, total);
}